// GCN_BiLSTM_nnatt_34368328302745
// MI455X (gfx1250) — compile-verified
//
#include <hip/hip_runtime.h>
#include <hip/hip_bf16.h>
#include <math.h>

// ---------------------------------------------------------------------------
// Types for WMMA fragments (CDNA5 / gfx1250, wave32)
// ---------------------------------------------------------------------------
typedef float v2f __attribute__((ext_vector_type(2)));
typedef float v8f __attribute__((ext_vector_type(8)));

// ---------------------------------------------------------------------------
// Tiled GEMM on V_WMMA_F32_16X16X4_F32 with async LDS staging.
//   C[m,n] = scale * sum_k A[m,k]*B(k,n)  (+ bias[n]) (relu optional)
//   TRANSB==1: B(k,n) = Bm[n*ldb + k]   (x @ W.T, W row-major N x K)
//   TRANSB==0: B(k,n) = Bm[k*ldb + n]   (row-major K x N)
//
// Block = 128 threads (4 waves, wave32). Macro tile 16 x 64; BK = 32.
// Wave w computes the 16x16 tile at column offset w*16.
// A and (TRANSB=1) B chunks are staged into LDS via the gfx1250 async path
// (global_load_async_to_lds_b128, ASYNCcnt), k-contiguous with rows padded to
// 36 floats (144B) so every 16B deposit is aligned and fragment b64 reads are
// bank-conflict free.  All call sites have M%16==0, N%64==0, K%32==0, so
// there is no edge handling and EXEC is all-ones for every WMMA.
//
// Fragment layouts per CDNA5 ISA 7.12.2:
//   A (16x4):  lanes 0-15 rows M=lane (K0,K1); lanes 16-31 (K2,K3)
//   B (4x16):  lanes 0-15 cols N=lane (K0,K1); lanes 16-31 (K2,K3)
//   C (16x16): VGPR r: lanes 0-15 -> M=r, N=lane; lanes 16-31 -> M=r+8
// ---------------------------------------------------------------------------
#define GEMM_BM 16
#define GEMM_BN 64
#define GEMM_BK 32
#define GEMM_LDSK 36   // padded row stride in floats (144 B, 16B-aligned rows)

template <int TRANSB, int RELU, int BIAS>
__global__ void wmma_gemm(const float* __restrict__ A, const float* __restrict__ Bm,
                          const float* __restrict__ bias, float* __restrict__ C,
                          int M, int N, int K, int lda, int ldb, int ldc, float scale,
                          long aBatch, long bBatch, long cBatch) {
    __shared__ float As[GEMM_BM * GEMM_LDSK];
    __shared__ float Bs[GEMM_BN * GEMM_LDSK];

    A  += (long)blockIdx.y * aBatch;
    Bm += (long)blockIdx.y * bBatch;
    C  += (long)blockIdx.y * cBatch;

    const int tilesN = N / GEMM_BN;
    const int bm = (blockIdx.x / tilesN) * GEMM_BM;
    const int bn = (blockIdx.x % tilesN) * GEMM_BN;

    const int tid  = threadIdx.x;        // 0..127
    const int wave = tid >> 5;           // 0..3
    const int lane = tid & 31;
    const int half = lane >> 4;          // 0 -> K{0,1}, 1 -> K{2,3}
    const int lmod = lane & 15;
    const int wn   = wave << 4;          // wave's column offset in macro tile

    // --- staging geometry (constant per thread) ---
    const int am = tid >> 3;             // 0..15   (A row)
    const int ak = (tid & 7) << 2;       // 0,4..28 (A k chunk)
    const uint32_t ldsA = (uint32_t)(uintptr_t)&As[am * GEMM_LDSK + ak];
    const float* agrow = A + (size_t)(bm + am) * lda + ak;

    // TRANSB=1 B staging: 64 rows x 32 k = 512 float4 chunks, 4 per thread.
    uint32_t ldsB[4];
    const float* bgrow[4];
    if (TRANSB) {
#pragma unroll
        for (int i = 0; i < 4; ++i) {
            const int cb = tid + (i << 7);       // 0..511
            const int n  = cb >> 3;              // 0..63
            const int kc = (cb & 7) << 2;        // 0..28
            ldsB[i]  = (uint32_t)(uintptr_t)&Bs[n * GEMM_LDSK + kc];
            bgrow[i] = Bm + (size_t)(bn + n) * ldb + kc;
        }
    }

    v8f acc = {};
    for (int k0 = 0; k0 < K; k0 += GEMM_BK) {
        // ---- stage A chunk (async copy global -> LDS) ----
        {
            const uint64_t ga = (uint64_t)(uintptr_t)(agrow + k0);
            asm volatile("global_load_async_to_lds_b128 %0, %1, off"
                         :: "v"(ldsA), "v"(ga) : "memory");
        }
        // ---- stage B chunk ----
        if (TRANSB) {
#pragma unroll
            for (int i = 0; i < 4; ++i) {
                const uint64_t gb = (uint64_t)(uintptr_t)(bgrow[i] + k0);
                asm volatile("global_load_async_to_lds_b128 %0, %1, off"
                             :: "v"(ldsB[i]), "v"(gb) : "memory");
            }
        } else {
            // Bs[n][k] = Bm[(k0+k)*ldb + bn + n]; coalesced in n.
#pragma unroll
            for (int i = 0; i < (GEMM_BK * GEMM_BN) / 128; ++i) {
                const int idx = tid + (i << 7);
                const int kk  = idx >> 6;        // 0..31
                const int n   = idx & 63;        // 0..63
                Bs[n * GEMM_LDSK + kk] = Bm[(size_t)(k0 + kk) * ldb + bn + n];
            }
        }
        asm volatile("s_wait_asynccnt 0x0" ::: "memory");
        __syncthreads();

        // ---- 8 WMMA steps from LDS ----
#pragma unroll
        for (int kk = 0; kk < GEMM_BK; kk += 4) {
            const int k2 = kk + half * 2;
            v2f a, b;
            a.x = As[lmod * GEMM_LDSK + k2];
            a.y = As[lmod * GEMM_LDSK + k2 + 1];
            b.x = Bs[(wn + lmod) * GEMM_LDSK + k2];
            b.y = Bs[(wn + lmod) * GEMM_LDSK + k2 + 1];
            acc = __builtin_amdgcn_wmma_f32_16x16x4_f32(
                false, a, false, b, (short)0, acc, false, false);
        }
        __syncthreads();
    }

#pragma unroll
    for (int r = 0; r < 8; ++r) {
        const int row = bm + half * 8 + r;
        const int col = bn + wn + lmod;
        float v = acc[r] * scale;
        if (BIAS) v += bias[col];
        if (RELU) v = fmaxf(v, 0.0f);
        C[(size_t)row * ldc + col] = v;
    }
}

// ---------------------------------------------------------------------------
// Layer-0 gate pre-activation: xg[t,j] = bih[j] + sum_{k<14} x[t,k]*W[j,k]
// ---------------------------------------------------------------------------
__global__ void xg_small(const float* __restrict__ x, const float* __restrict__ W,
                         const float* __restrict__ bih, float* __restrict__ xg,
                         int T, int din, int G) {
    int idx = blockIdx.x * blockDim.x + threadIdx.x;
    if (idx >= T * G) return;
    int t = idx / G, j = idx % G;
    const float* xr = x + (size_t)t * din;
    const float* wr = W + (size_t)j * din;
    float acc = bih[j];
    for (int k = 0; k < din; ++k) acc += xr[k] * wr[k];
    xg[idx] = acc;
}

// ---------------------------------------------------------------------------
// LSTM sequential scan. One block per direction (gridDim.x == 2), 512 threads.
// ---------------------------------------------------------------------------
struct LstmArgs {
    const float* xg[2];
    const float* Whh[2];
    const float* bhh[2];
    float* out;
    int T;
    int outStride;
    int outOff[2];
    int rev[2];
};

__device__ __forceinline__ float sigm(float x) { return 1.0f / (1.0f + __expf(-x)); }

__global__ void lstm_scan(LstmArgs a) {
    const int dir = blockIdx.x;
    const float* xg  = a.xg[dir];
    const float* Whh = a.Whh[dir];
    const float* bhh = a.bhh[dir];
    const int rev    = a.rev[dir];
    const int off    = a.outOff[dir];
    const int T      = a.T;

    __shared__ float sh[128];
    __shared__ float sc[128];
    __shared__ float sg[512];

    const int tid = threadIdx.x;          // 0..511
    if (tid < 128) { sh[tid] = 0.0f; sc[tid] = 0.0f; }
    __syncthreads();

    const float* wr = Whh + (size_t)tid * 128;
    const float  bh = bhh[tid];

    for (int t = 0; t < T; ++t) {
        const int tt = rev ? (T - 1 - t) : t;
        float acc = xg[(size_t)tt * 512 + tid] + bh;
#pragma unroll 4
        for (int k = 0; k < 128; ++k) acc += wr[k] * sh[k];
        sg[tid] = acc;
        __syncthreads();
        if (tid < 128) {
            const float i  = sigm(sg[tid]);
            const float f  = sigm(sg[128 + tid]);
            const float gg = tanhf(sg[256 + tid]);
            const float o  = sigm(sg[384 + tid]);
            const float c  = f * sc[tid] + i * gg;
            sc[tid] = c;
            const float h = o * tanhf(c);
            sh[tid] = h;
            a.out[(size_t)tt * a.outStride + off + tid] = h;
        }
        __syncthreads();
    }
}

// ---------------------------------------------------------------------------
// Row softmax (in place). One block (256 threads) per row of length L.
// ---------------------------------------------------------------------------
__global__ void softmax_rows(float* __restrict__ s, int L) {
    float* r = s + (size_t)blockIdx.x * L;
    __shared__ float red[256];
    const int tid = threadIdx.x;

    float m = -3.4e38f;
    for (int i = tid; i < L; i += 256) m = fmaxf(m, r[i]);
    red[tid] = m; __syncthreads();
    for (int o = 128; o > 0; o >>= 1) { if (tid < o) red[tid] = fmaxf(red[tid], red[tid + o]); __syncthreads(); }
    m = red[0]; __syncthreads();

    float sum = 0.0f;
    for (int i = tid; i < L; i += 256) { float e = __expf(r[i] - m); r[i] = e; sum += e; }
    red[tid] = sum; __syncthreads();
    for (int o = 128; o > 0; o >>= 1) { if (tid < o) red[tid] += red[tid + o]; __syncthreads(); }
    const float inv = 1.0f / red[0];
    for (int i = tid; i < L; i += 256) r[i] *= inv;
}

// Mean over 4 heads: out[qk] = 0.25 * sum_h attn[h][qk]
__global__ void attn_mean(const float* __restrict__ attn, float* __restrict__ out, int QK) {
    int i = blockIdx.x * blockDim.x + threadIdx.x;
    if (i >= QK) return;
    out[i] = 0.25f * (attn[i] + attn[i + QK] + attn[i + 2 * QK] + attn[i + 3 * QK]);
}

// ---------------------------------------------------------------------------
// Graph segment-sum: agg[dst[e], :] += x[src[e], :]   (F = 128, float4 lanes)
// ---------------------------------------------------------------------------
__global__ void zero_buf(float* __restrict__ p, long n) {
    long i = (long)blockIdx.x * blockDim.x + threadIdx.x;
    if (i < n) p[i] = 0.0f;
}

__global__ void seg_gather(const int* __restrict__ src, const int* __restrict__ dst,
                           const float* __restrict__ x, float* __restrict__ agg,
                           long E, int F) {
    const int cpe = F >> 2;                      // float4 chunks per edge
    long tid = (long)blockIdx.x * blockDim.x + threadIdx.x;
    if (tid >= E * cpe) return;
    const long e = tid / cpe;
    const int  c = (int)(tid % cpe) << 2;
    const int  s = src[e];
    const int  d = dst[e];
    const float4 v = *(const float4*)(x + (size_t)s * F + c);
    float* ap = agg + (size_t)d * F + c;
    atomicAdd(ap + 0, v.x);
    atomicAdd(ap + 1, v.y);
    atomicAdd(ap + 2, v.z);
    atomicAdd(ap + 3, v.w);
}

// Column mean of an Nrows x Ncols matrix; one block per column.
__global__ void col_mean(const float* __restrict__ h, float* __restrict__ out,
                         int Nrows, int Ncols) {
    const int col = blockIdx.x;
    __shared__ float red[256];
    float s = 0.0f;
    for (int r = threadIdx.x; r < Nrows; r += 256) s += h[(size_t)r * Ncols + col];
    red[threadIdx.x] = s; __syncthreads();
    for (int o = 128; o > 0; o >>= 1) { if (threadIdx.x < o) red[threadIdx.x] += red[threadIdx.x + o]; __syncthreads(); }
    if (threadIdx.x == 0) out[col] = red[0] / (float)Nrows;
}

// feats = concat(hg[256], attn_out[511,:256], smiles[638]) -> 1150
__global__ void concat_feats(const float* __restrict__ hg, const float* __restrict__ attn_out,
                             const float* __restrict__ smiles, float* __restrict__ feats) {
    int i = blockIdx.x * blockDim.x + threadIdx.x;
    if (i < 256)       feats[i] = hg[i];
    else if (i < 512)  feats[i] = attn_out[(size_t)511 * 256 + (i - 256)];
    else if (i < 1150) feats[i] = smiles[i - 512];
}

// y[n] = (relu?) b[n] + dot(W[n,:K], x)
__global__ void matvec(const float* __restrict__ W, const float* __restrict__ b,
                       const float* __restrict__ x, float* __restrict__ y,
                       int N, int K, int relu) {
    int n = blockIdx.x * blockDim.x + threadIdx.x;
    if (n >= N) return;
    const float* wr = W + (size_t)n * K;
    float acc = b[n];
    for (int k = 0; k < K; ++k) acc += wr[k] * x[k];
    if (relu) acc = fmaxf(acc, 0.0f);
    y[n] = acc;
}

// ---------------------------------------------------------------------------
// Host orchestration
// ---------------------------------------------------------------------------
typedef void (*GemmFn)(const float*, const float*, const float*, float*,
                       int, int, int, int, int, int, float, long, long, long);

extern "C" void kernel_launch(void* const* d_in, const int* in_sizes, int n_in,
                              void* d_out, int out_size, void* d_ws, size_t ws_size,
                              hipStream_t stream) {
    const int H = 128, E = 256, NH = 4, L = 512;
    const int N_NODES = 50000;
    const long EDGES = 600000;

    int p = 0;
    const float* g_mol      = (const float*)d_in[p++];
    const float* feat_seq   = (const float*)d_in[p++];
    const float* smiles     = (const float*)d_in[p++];
    const float* l0f_Wih = (const float*)d_in[p++]; const float* l0f_Whh = (const float*)d_in[p++];
    const float* l0f_bih = (const float*)d_in[p++]; const float* l0f_bhh = (const float*)d_in[p++];
    const float* l0b_Wih = (const float*)d_in[p++]; const float* l0b_Whh = (const float*)d_in[p++];
    const float* l0b_bih = (const float*)d_in[p++]; const float* l0b_bhh = (const float*)d_in[p++];
    const float* l1f_Wih = (const float*)d_in[p++]; const float* l1f_Whh = (const float*)d_in[p++];
    const float* l1f_bih = (const float*)d_in[p++]; const float* l1f_bhh = (const float*)d_in[p++];
    const float* l1b_Wih = (const float*)d_in[p++]; const float* l1b_Whh = (const float*)d_in[p++];
    const float* l1b_bih = (const float*)d_in[p++]; const float* l1b_bhh = (const float*)d_in[p++];
    const float* fc_w  = (const float*)d_in[p++]; const float* fc_b  = (const float*)d_in[p++];
    const float* mha_in_w  = (const float*)d_in[p++]; const float* mha_in_b  = (const float*)d_in[p++];
    const float* mha_out_w = (const float*)d_in[p++]; const float* mha_out_b = (const float*)d_in[p++];
    const float* gc1_w = (const float*)d_in[p++]; const float* gc1_b = (const float*)d_in[p++];
    const float* gc2_w = (const float*)d_in[p++]; const float* gc2_b = (const float*)d_in[p++];
    const float* fc1_w = (const float*)d_in[p++]; const float* fc1_b = (const float*)d_in[p++];
    const float* fc2_w = (const float*)d_in[p++]; const float* fc2_b = (const float*)d_in[p++];
    const float* fc3_w = (const float*)d_in[p++]; const float* fc3_b = (const float*)d_in[p++];
    const float* fc4_w = (const float*)d_in[p++]; const float* fc4_b = (const float*)d_in[p++];
    const int* src = (const int*)d_in[p++];
    const int* dst = (const int*)d_in[p++];

    float* out = (float*)d_out;   // out[0] = o ; out[1..] = attn mean (512*512)

    // ---- workspace carve-out (floats) ----
    float* ws = (float*)d_ws;
    size_t off = 0;
    auto alloc = [&](size_t n) { float* q = ws + off; off += n; return q; };
    float* xg0f = alloc((size_t)L * 512);
    float* xg0b = alloc((size_t)L * 512);
    float* hs0  = alloc((size_t)L * 256);
    float* xg1f = alloc((size_t)L * 512);
    float* xg1b = alloc((size_t)L * 512);
    float* hs1  = alloc((size_t)L * 256);
    float* proj = alloc((size_t)L * 256);
    float* qkv  = alloc((size_t)L * 768);
    float* attn = alloc((size_t)NH * L * L);
    float* attnv= alloc((size_t)L * 256);
    float* attno= alloc((size_t)L * 256);
    float* agg  = alloc((size_t)N_NODES * 128);
    float* h1   = alloc((size_t)N_NODES * 128);
    float* h2   = alloc((size_t)N_NODES * 256);
    float* hg   = alloc(256);
    float* feats= alloc(1152);
    float* m1   = alloc(576);
    float* m2   = alloc(256);
    float* m3   = alloc(64);
    (void)ws_size; (void)n_in; (void)in_sizes; (void)out_size; (void)H; (void)E;

    static const GemmFn gemm_tbl[8] = {
        wmma_gemm<0,0,0>, wmma_gemm<0,0,1>, wmma_gemm<0,1,0>, wmma_gemm<0,1,1>,
        wmma_gemm<1,0,0>, wmma_gemm<1,0,1>, wmma_gemm<1,1,0>, wmma_gemm<1,1,1>};

    auto gemm = [&](const float* A, const float* Bm, const float* bias, float* C,
                    int M, int N, int K, int lda, int ldb, int ldc,
                    int transB, int relu, float scale,
                    long aB, long bB, long cB, int batch) {
        GemmFn fn = gemm_tbl[transB * 4 + relu * 2 + (bias != nullptr)];
        dim3 grid((M / GEMM_BM) * (N / GEMM_BN), batch);
        hipLaunchKernelGGL(fn, grid, dim3(128), 0, stream,
                           A, Bm, bias, C, M, N, K, lda, ldb, ldc, scale, aB, bB, cB);
    };

    // ---- BiLSTM layer 0 ----
    hipLaunchKernelGGL(xg_small, dim3((L * 512 + 255) / 256), dim3(256), 0, stream,
                       g_mol, l0f_Wih, l0f_bih, xg0f, L, 14, 512);
    hipLaunchKernelGGL(xg_small, dim3((L * 512 + 255) / 256), dim3(256), 0, stream,
                       g_mol, l0b_Wih, l0b_bih, xg0b, L, 14, 512);
    {
        LstmArgs a;
        a.xg[0] = xg0f; a.xg[1] = xg0b;
        a.Whh[0] = l0f_Whh; a.Whh[1] = l0b_Whh;
        a.bhh[0] = l0f_bhh; a.bhh[1] = l0b_bhh;
        a.out = hs0; a.T = L; a.outStride = 256;
        a.outOff[0] = 0; a.outOff[1] = 128; a.rev[0] = 0; a.rev[1] = 1;
        hipLaunchKernelGGL(lstm_scan, dim3(2), dim3(512), 0, stream, a);
    }

    // ---- BiLSTM layer 1 : gate pre-activations via WMMA, then scan ----
    gemm(hs0, l1f_Wih, l1f_bih, xg1f, L, 512, 256, 256, 256, 512, 1, 0, 1.0f, 0, 0, 0, 1);
    gemm(hs0, l1b_Wih, l1b_bih, xg1b, L, 512, 256, 256, 256, 512, 1, 0, 1.0f, 0, 0, 0, 1);
    {
        LstmArgs a;
        a.xg[0] = xg1f; a.xg[1] = xg1b;
        a.Whh[0] = l1f_Whh; a.Whh[1] = l1b_Whh;
        a.bhh[0] = l1f_bhh; a.bhh[1] = l1b_bhh;
        a.out = hs1; a.T = L; a.outStride = 256;
        a.outOff[0] = 0; a.outOff[1] = 128; a.rev[0] = 0; a.rev[1] = 1;
        hipLaunchKernelGGL(lstm_scan, dim3(2), dim3(512), 0, stream, a);
    }

    // ---- projection + MHA ----
    gemm(hs1, fc_w, fc_b, proj, L, 256, 256, 256, 256, 256, 1, 0, 1.0f, 0, 0, 0, 1);
    gemm(proj, mha_in_w, mha_in_b, qkv, L, 768, 256, 256, 256, 768, 1, 0, 1.0f, 0, 0, 0, 1);
    // scores[h] = (Q_h @ K_h^T) / 8 ; batch over 4 heads
    gemm(qkv, qkv + 256, nullptr, attn, L, L, 64, 768, 768, L,
         1, 0, 0.125f, 64, 64, (long)L * L, NH);
    hipLaunchKernelGGL(softmax_rows, dim3(NH * L), dim3(256), 0, stream, attn, L);
    hipLaunchKernelGGL(attn_mean, dim3((L * L + 255) / 256), dim3(256), 0, stream,
                       attn, out + 1, L * L);
    // per-head attn @ V, heads write disjoint 64-wide column slices
    gemm(attn, qkv + 512, nullptr, attnv, L, 64, L, L, 768, 256,
         0, 0, 1.0f, (long)L * L, 64, 64, NH);
    gemm(attnv, mha_out_w, mha_out_b, attno, L, 256, 256, 256, 256, 256, 1, 0, 1.0f, 0, 0, 0, 1);

    // ---- GCN ----
    const long aggN = (long)N_NODES * 128;
    hipLaunchKernelGGL(zero_buf, dim3((unsigned)((aggN + 255) / 256)), dim3(256), 0, stream, agg, aggN);
    hipLaunchKernelGGL(seg_gather, dim3((unsigned)((EDGES * 32 + 255) / 256)), dim3(256), 0, stream,
                       src, dst, feat_seq, agg, EDGES, 128);
    gemm(agg, gc1_w, gc1_b, h1, N_NODES, 128, 128, 128, 128, 128, 0, 1, 1.0f, 0, 0, 0, 1);
    hipLaunchKernelGGL(zero_buf, dim3((unsigned)((aggN + 255) / 256)), dim3(256), 0, stream, agg, aggN);
    hipLaunchKernelGGL(seg_gather, dim3((unsigned)((EDGES * 32 + 255) / 256)), dim3(256), 0, stream,
                       src, dst, h1, agg, EDGES, 128);
    gemm(agg, gc2_w, gc2_b, h2, N_NODES, 256, 128, 128, 256, 256, 0, 0, 1.0f, 0, 0, 0, 1);
    hipLaunchKernelGGL(col_mean, dim3(256), dim3(256), 0, stream, h2, hg, N_NODES, 256);

    // ---- MLP head ----
    hipLaunchKernelGGL(concat_feats, dim3(5), dim3(256), 0, stream, hg, attno, smiles, feats);
    hipLaunchKernelGGL(matvec, dim3((575 + 255) / 256), dim3(256), 0, stream, fc1_w, fc1_b, feats, m1, 575, 1150, 1);
    hipLaunchKernelGGL(matvec, dim3(1), dim3(256), 0, stream, fc2_w, fc2_b, m1, m2, 256, 575, 1);
    hipLaunchKernelGGL(matvec, dim3(1), dim3(256), 0, stream, fc3_w, fc3_b, m2, m3, 64, 256, 1);
    hipLaunchKernelGGL(matvec, dim3(1), dim3(256), 0, stream, fc4_w, fc4_b, m3, out, 1, 64, 0);
}